// graphSAGE_ENCODER_85461259255857
// MI455X (gfx1250) — compile-verified
//
#include <hip/hip_runtime.h>

typedef __attribute__((ext_vector_type(2))) float v2f;
typedef __attribute__((ext_vector_type(8))) float v8f;

#define DIM 128
#define NPAP 100000
#define NAUT 50000

// ---------------- utility kernels ----------------

__global__ void k_zero(float* __restrict__ p, long n) {
  long i = (long)blockIdx.x * blockDim.x + threadIdx.x;
  if (i < n) p[i] = 0.0f;
}

// one wave (32 lanes) per edge: agg[dst] += X[src]; cnt[dst] += 1
__global__ void k_scatter(const float* __restrict__ X,
                          const int* __restrict__ src,
                          const int* __restrict__ dst,
                          float* __restrict__ agg,
                          float* __restrict__ cnt, int E) {
  int t = blockIdx.x * blockDim.x + threadIdx.x;
  int e = t >> 5;
  int lane = t & 31;
  if (e >= E) return;
  int s = src[e];
  int d = dst[e];
  const float* xs = X + (long)s * DIM;
  float* ad = agg + (long)d * DIM;
#pragma unroll
  for (int j = 0; j < 4; ++j) {
    unsafeAtomicAdd(&ad[lane + 32 * j], xs[lane + 32 * j]);
  }
  if (lane == 0) unsafeAtomicAdd(&cnt[d], 1.0f);
}

__global__ void k_recip(float* __restrict__ cnt, int n) {
  int i = blockIdx.x * blockDim.x + threadIdx.x;
  if (i < n) cnt[i] = 1.0f / fmaxf(cnt[i], 1.0f);
}

__global__ void k_addw(const float* __restrict__ a, const float* __restrict__ b,
                       float* __restrict__ o) {
  int i = blockIdx.x * blockDim.x + threadIdx.x;
  if (i < DIM * DIM) o[i] = a[i] + b[i];
}

// ---------------- WMMA GEMM ----------------
// C[M,128] = (accum ? C : 0) + bias + (rowscale .* A[M,128]) @ B[128,128]
// optional relu epilogue. One wave computes a 16x128 strip.
// Uses V_WMMA_F32_16X16X4_F32 (fp32 in/out, K=4), B staged in 64KB LDS.
__global__ __launch_bounds__(128) void k_gemm(
    const float* __restrict__ A, const float* __restrict__ rowscale,
    const float* __restrict__ B, const float* __restrict__ bias,
    float* __restrict__ C, int M, int accum, int relu) {
  __shared__ float Bs[DIM * DIM];
  for (int i = threadIdx.x; i < DIM * DIM; i += blockDim.x) Bs[i] = B[i];
  __syncthreads();

  int wave = threadIdx.x >> 5;
  int lane = threadIdx.x & 31;
  int m0 = blockIdx.x * 64 + wave * 16;
  if (m0 >= M) return;  // wave-uniform: EXEC stays all-ones for WMMA

  int hi = lane >> 4;   // 0 or 1 (lane half)
  int lo = lane & 15;

  // C/D 16x16 f32 layout: VGPR r holds M = m0 + hi*8 + r, N = lo (per n-tile)
  v8f acc[8];
#pragma unroll
  for (int nt = 0; nt < 8; ++nt) {
#pragma unroll
    for (int r = 0; r < 8; ++r) {
      int row = m0 + hi * 8 + r;
      int col = nt * 16 + lo;
      float c = accum ? C[(long)row * DIM + col] : 0.0f;
      if (bias) c += bias[col];
      acc[nt][r] = c;
    }
  }

  // A 16x4 f32 layout: lane (hi,lo) holds A[m0+lo][k0 + hi*2 + {0,1}]
  int arow = m0 + lo;
  float s = rowscale ? rowscale[arow] : 1.0f;
  const float* Arow = A + (long)arow * DIM + hi * 2;

  for (int k0 = 0; k0 < DIM; k0 += 4) {
    v2f a;
    a.x = Arow[k0] * s;
    a.y = Arow[k0 + 1] * s;
    int kb = k0 + hi * 2;
#pragma unroll
    for (int nt = 0; nt < 8; ++nt) {
      // B 4x16 f32 layout: lane (hi,lo) holds B[kb+{0,1}][nt*16+lo]
      v2f b;
      b.x = Bs[kb * DIM + nt * 16 + lo];
      b.y = Bs[(kb + 1) * DIM + nt * 16 + lo];
      acc[nt] = __builtin_amdgcn_wmma_f32_16x16x4_f32(
          false, a, false, b, (short)0, acc[nt], false, false);
    }
  }

#pragma unroll
  for (int nt = 0; nt < 8; ++nt) {
#pragma unroll
    for (int r = 0; r < 8; ++r) {
      int row = m0 + hi * 8 + r;
      int col = nt * 16 + lo;
      float v = acc[nt][r];
      if (relu) v = fmaxf(v, 0.0f);
      C[(long)row * DIM + col] = v;
    }
  }
}

// ---------------- host orchestration ----------------

static inline void zero_n(float* p, long n, hipStream_t st) {
  k_zero<<<(unsigned)((n + 255) / 256), 256, 0, st>>>(p, n);
}
static inline void scatter(const float* X, const int* s, const int* d,
                           float* agg, float* cnt, int E, hipStream_t st) {
  long threads = (long)E * 32;
  k_scatter<<<(unsigned)((threads + 255) / 256), 256, 0, st>>>(X, s, d, agg,
                                                              cnt, E);
}
static inline void recip(float* cnt, int n, hipStream_t st) {
  k_recip<<<(unsigned)((n + 255) / 256), 256, 0, st>>>(cnt, n);
}
static inline void gemm(const float* A, const float* rs, const float* B,
                        const float* bias, float* C, int M, int accum, int relu,
                        hipStream_t st) {
  k_gemm<<<(unsigned)((M + 63) / 64), 128, 0, st>>>(A, rs, B, bias, C, M,
                                                   accum, relu);
}

extern "C" void kernel_launch(void* const* d_in, const int* in_sizes, int n_in,
                              void* d_out, int out_size, void* d_ws,
                              size_t ws_size, hipStream_t stream) {
  const float* xp = (const float*)d_in[0];
  const float* xa = (const float*)d_in[1];
  const float* Wm1 = (const float*)d_in[2];
  const float* b1 = (const float*)d_in[3];
  const float* Wr1 = (const float*)d_in[4];
  const float* Wm2 = (const float*)d_in[5];
  const float* b2 = (const float*)d_in[6];
  const float* Wr2 = (const float*)d_in[7];
  const int* ci_s = (const int*)d_in[8];
  const int* ci_d = (const int*)d_in[9];
  const int* wr_s = (const int*)d_in[10];
  const int* wr_d = (const int*)d_in[11];
  const int* wb_s = (const int*)d_in[12];
  const int* wb_d = (const int*)d_in[13];
  int E_c = in_sizes[8];
  int E_w = in_sizes[10];
  int E_b = in_sizes[12];

  // workspace layout (floats)
  float* ws = (float*)d_ws;
  float* agg = ws;                         // NPAP*128 (reused, NA fits too)
  float* cnt = ws + 12800000;              // NPAP counts
  float* wrc = ws + 12900000;              // 128*128 combined root weight
  float* hp = ws + 12920000;               // NPAP*128 hidden paper
  float* ha = hp + (long)NPAP * DIM;       // NAUT*128 hidden author

  float* op = (float*)d_out;               // [NPAP,128]
  float* oa = op + (long)NPAP * DIM;       // [NAUT,128]

  const long WW = (long)DIM * DIM;

  // ---------------- layer 1 -> hp, ha (with ReLU) ----------------
  // paper output: sage(cites: p->p) + sage(writes: a->p)
  zero_n(agg, (long)NPAP * DIM, stream);
  zero_n(cnt, NPAP, stream);
  scatter(xp, ci_s, ci_d, agg, cnt, E_c, stream);
  recip(cnt, NPAP, stream);
  gemm(agg, cnt, Wm1 + 0 * WW, b1 + 0 * DIM, hp, NPAP, 0, 0, stream);

  zero_n(agg, (long)NPAP * DIM, stream);
  zero_n(cnt, NPAP, stream);
  scatter(xa, wr_s, wr_d, agg, cnt, E_w, stream);
  recip(cnt, NPAP, stream);
  gemm(agg, cnt, Wm1 + 1 * WW, b1 + 1 * DIM, hp, NPAP, 1, 0, stream);

  k_addw<<<(DIM * DIM + 255) / 256, 256, 0, stream>>>(Wr1 + 0 * WW,
                                                      Wr1 + 1 * WW, wrc);
  gemm(xp, nullptr, wrc, nullptr, hp, NPAP, 1, 1, stream);  // relu

  // author output: sage(wb: p->a)
  zero_n(agg, (long)NAUT * DIM, stream);
  zero_n(cnt, NAUT, stream);
  scatter(xp, wb_s, wb_d, agg, cnt, E_b, stream);
  recip(cnt, NAUT, stream);
  gemm(agg, cnt, Wm1 + 2 * WW, b1 + 2 * DIM, ha, NAUT, 0, 0, stream);
  gemm(xa, nullptr, Wr1 + 2 * WW, nullptr, ha, NAUT, 1, 1, stream);  // relu

  // ---------------- layer 2 -> op, oa ----------------
  zero_n(agg, (long)NPAP * DIM, stream);
  zero_n(cnt, NPAP, stream);
  scatter(hp, ci_s, ci_d, agg, cnt, E_c, stream);
  recip(cnt, NPAP, stream);
  gemm(agg, cnt, Wm2 + 0 * WW, b2 + 0 * DIM, op, NPAP, 0, 0, stream);

  zero_n(agg, (long)NPAP * DIM, stream);
  zero_n(cnt, NPAP, stream);
  scatter(ha, wr_s, wr_d, agg, cnt, E_w, stream);
  recip(cnt, NPAP, stream);
  gemm(agg, cnt, Wm2 + 1 * WW, b2 + 1 * DIM, op, NPAP, 1, 0, stream);

  k_addw<<<(DIM * DIM + 255) / 256, 256, 0, stream>>>(Wr2 + 0 * WW,
                                                      Wr2 + 1 * WW, wrc);
  gemm(hp, nullptr, wrc, nullptr, op, NPAP, 1, 0, stream);

  zero_n(agg, (long)NAUT * DIM, stream);
  zero_n(cnt, NAUT, stream);
  scatter(hp, wb_s, wb_d, agg, cnt, E_b, stream);
  recip(cnt, NAUT, stream);
  gemm(agg, cnt, Wm2 + 2 * WW, b2 + 2 * DIM, oa, NAUT, 0, 0, stream);
  gemm(ha, nullptr, Wr2 + 2 * WW, nullptr, oa, NAUT, 1, 0, stream);
}